// HBV_57208964383105
// MI455X (gfx1250) — compile-verified
//
#include <hip/hip_runtime.h>
#include <stdint.h>

#define NSTEP 365
#define NGRID 1000
#define MU    16
#define NPAR  14
#define PRECSf 1e-5f

// ---- CDNA5 async global->LDS (ASYNCcnt-tracked DMA), inline asm ------------
__device__ __forceinline__ void async_b32(uint32_t ldsOff, const float* gp) {
  // LDS byte address from the ldsOff VGPR; global address from a 64-bit VGPR pair, no saddr
  asm volatile("global_load_async_to_lds_b32 %0, %1, off"
               :: "v"(ldsOff), "v"(gp) : "memory");
}
__device__ __forceinline__ void wait_async_le3() {
  asm volatile("s_wait_asynccnt 3" ::: "memory");
}
__device__ __forceinline__ void wait_async_0() {
  asm volatile("s_wait_asynccnt 0" ::: "memory");
}
__device__ __forceinline__ void wait_ds_0() {
  asm volatile("s_wait_dscnt 0" ::: "memory");
}

// a > 0 always in this model (SM clamped to >= 1e-5).
// Raw v_log_f32 / v_exp_f32 (base-2) -> 3 TRANS/VALU ops per pow.
__device__ __forceinline__ float fpow(float a, float b) {
#if __has_builtin(__builtin_amdgcn_logf) && __has_builtin(__builtin_amdgcn_exp2f)
  return __builtin_amdgcn_exp2f(b * __builtin_amdgcn_logf(a));
#else
  return __expf(b * __logf(a));
#endif
}

__global__ __launch_bounds__(32) void hbv_scan_kernel(
    const float* __restrict__ x,      // (NSTEP, NGRID, 3)
    const float* __restrict__ par,    // (NSTEP, NGRID, NPAR, MU)
    const int*   __restrict__ staind_p,
    float*       __restrict__ out)    // (NSTEP, NGRID, 1)
{
  // One wave32 per block: 2 grid cells x 16 mu members.
  __shared__ float sx [2][8];        // [buf][grid*3 + c]  (padded to 8)
  __shared__ float sb [2][2][MU];    // raw BETA   (param 0)
  __shared__ float sbe[2][2][MU];    // raw BETAET (param 12)

  const int lane = threadIdx.x;
  const int tx   = lane & (MU - 1);   // mu
  const int gs   = lane >> 4;         // 0 or 1
  const int g0   = blockIdx.x * 2;
  const int g    = g0 + gs;
  const int staind = *staind_p;

  const float LO[NPAR] = {1.0f,50.0f,0.05f,0.01f,0.001f,0.2f,0.0f,0.0f,-2.5f,0.5f,0.0f,0.0f,0.3f,0.0f};
  const float HI[NPAR] = {6.0f,1000.0f,0.9f,0.5f,0.2f,1.0f,10.0f,100.0f,2.5f,10.0f,0.1f,0.2f,5.0f,1.0f};

  // Static (time-frozen) parameters taken from step `staind`.
  const float* ps = par + (((size_t)staind * NGRID + g) * NPAR) * MU + tx;
  float tp[NPAR];
#pragma unroll
  for (int p = 0; p < NPAR; ++p)
    tp[p] = LO[p] + ps[p * MU] * (HI[p] - LO[p]);
  const float FC = tp[1], K0 = tp[2], K1 = tp[3], K2 = tp[4], LP = tp[5],
              PERCp = tp[6], UZL = tp[7], TT = tp[8], CFMAX = tp[9],
              CFR = tp[10], CWH = tp[11], C = tp[13];

  // LDS byte offsets (low 32 bits of a flat LDS pointer == workgroup-relative LDS addr)
  const uint32_t sx_o  = (uint32_t)(uintptr_t)&sx[0][0];
  const uint32_t sb_o  = (uint32_t)(uintptr_t)&sb[0][0][0];
  const uint32_t sbe_o = (uint32_t)(uintptr_t)&sbe[0][0][0];

  float SP = 0.001f, MW = 0.001f, SM = 0.001f, SUZ = 0.001f, SLZ = 0.001f;

  // Per-step strides (elements) for running pointers — no per-iter 64-bit mads.
  const size_t XSTP = (size_t)NGRID * 3;
  const size_t PSTP = (size_t)NGRID * NPAR * MU;

  // Prefetch one step into LDS buffer nb: exactly 3 async instructions per wave.
  auto issue = [&](const float* xstep, const float* pstep, int nb) {
    if (lane < 6)
      async_b32(sx_o + (uint32_t)(nb * 8 + lane) * 4u, xstep + lane);  // 6 contiguous floats
    const uint32_t po = (uint32_t)((nb * 2 + gs) * MU + tx) * 4u;
    async_b32(sb_o  + po, pstep);            // BETA   row (param 0)  : 64B coalesced
    async_b32(sbe_o + po, pstep + 12 * MU);  // BETAET row (param 12) : 64B coalesced
  };

  const float* xn = x   + (size_t)g0 * 3;                       // step-t x ptr (lane-invariant base)
  const float* pn = par + ((size_t)g * NPAR) * MU + tx;         // step-t dyn-param ptr
  float*       op = out + g;

  issue(xn, pn, 0);

#pragma unroll 2
  for (int t = 0; t < NSTEP; ++t) {
    const int b = t & 1;
    if (t + 1 < NSTEP) {
      xn += XSTP; pn += PSTP;
      wait_ds_0();            // prior-iter LDS reads retired before DMA overwrites buffer b^1
      issue(xn, pn, b ^ 1);   // +3 outstanding (max 6 in flight)
      wait_async_le3();       // async completes in order -> buffer b's 3 are done
    } else {
      wait_async_0();
    }

    const float P    = sx[b][gs * 3 + 0];
    const float T    = sx[b][gs * 3 + 1];
    const float ETp  = sx[b][gs * 3 + 2];
    const float BETA   = 1.0f + sb [b][gs][tx] * (6.0f - 1.0f);
    const float BETAET = 0.3f + sbe[b][gs][tx] * (5.0f - 0.3f);

    const float RAIN = (T >= TT) ? P : 0.0f;
    const float SNOW = (T <  TT) ? P : 0.0f;
    SP += SNOW;
    const float melt = fminf(fmaxf(CFMAX * (T - TT), 0.0f), SP);
    MW += melt;  SP -= melt;
    const float refr = fminf(fmaxf(CFR * CFMAX * (TT - T), 0.0f), MW);
    SP += refr;  MW -= refr;
    const float tosoil = fmaxf(MW - CWH * SP, 0.0f);
    MW -= tosoil;
    const float wet = fminf(fmaxf(fpow(SM / FC, BETA), 0.0f), 1.0f);
    const float recharge = (RAIN + tosoil) * wet;
    const float excess = fmaxf(SM - FC, 0.0f);
    SM -= excess;
    const float evapf = fminf(fmaxf(fpow(SM / (LP * FC), BETAET), 0.0f), 1.0f);
    const float ETact = fminf(SM, ETp * evapf);
    SM = fmaxf(SM - ETact, PRECSf);
    SM = SM + RAIN + tosoil - recharge;
    const float cap = fminf(SLZ, C * SLZ * (1.0f - fminf(SM / FC, 1.0f)));
    SM  = fmaxf(SM + cap, PRECSf);
    SLZ = fmaxf(SLZ - cap, PRECSf);
    SUZ = SUZ + recharge + excess;
    const float PERC = fminf(SUZ, PERCp);
    SUZ -= PERC;
    const float Q0 = K0 * fmaxf(SUZ - UZL, 0.0f);
    SUZ -= Q0;
    const float Q1 = K1 * SUZ;
    SUZ -= Q1;
    SLZ += PERC;
    const float Q2 = K2 * SLZ;
    SLZ -= Q2;

    // mean over 16 mu lanes (xor tree stays inside each 16-lane half of the wave32)
    float q = Q0 + Q1 + Q2;
    q += __shfl_xor(q, 1);
    q += __shfl_xor(q, 2);
    q += __shfl_xor(q, 4);
    q += __shfl_xor(q, 8);
    if (tx == 0)
      *op = q * (1.0f / 16.0f);
    op += NGRID;
  }
}

extern "C" void kernel_launch(void* const* d_in, const int* in_sizes, int n_in,
                              void* d_out, int out_size, void* d_ws, size_t ws_size,
                              hipStream_t stream) {
  (void)in_sizes; (void)n_in; (void)d_ws; (void)ws_size; (void)out_size;
  const float* x      = (const float*)d_in[0];
  const float* par    = (const float*)d_in[1];
  const int*   staind = (const int*)d_in[2];
  float*       out    = (float*)d_out;

  dim3 grid(NGRID / 2);   // 500 single-wave workgroups: 2 grids x 16 mu each
  dim3 block(32);
  hipLaunchKernelGGL(hbv_scan_kernel, grid, block, 0, stream, x, par, staind, out);
}